// Debias_5265629905344
// MI455X (gfx1250) — compile-verified
//
#include <hip/hip_runtime.h>

// Debias.generate_exp_adj for MI455X (gfx1250, wave32).
//
// Cost model: dominated by streaming pred (1e6 x 51 f32 = 204 MB) -> ~9us at
// 23.3 TB/s. Strategy: double-buffered GLOBAL_LOAD_ASYNC_TO_LDS_B128 staging
// (ASYNCcnt-pipelined: load tile k+1 while computing tile k), per-block LDS
// histogram (ds_add_u32), one global-atomic merge per block; tiny epilogue
// does row sums via V_WMMA_F32_16X16X4_F32 (counts < 2^24 -> exact in f32).

#define NUM_REL 51
#define THREADS 128
#define TILE_ROWS 128
#define TILE_FLOATS (TILE_ROWS * NUM_REL)     // 6528 floats = 26112 B
#define TILE_F4 (TILE_FLOATS / 4)             // 1632 float4 per tile
#define ASYNC_ITERS 13                        // ceil(1632 / 128) -> uniform issue count
#define TILE_SLOTS (ASYNC_ITERS * THREADS)    // 1664 float4 (padded)
#define HIST_BINS (NUM_REL * NUM_REL)         // 2601
#define PAD_COLS 52                           // padded row stride for WMMA epilogue

typedef __attribute__((ext_vector_type(2))) float v2f;
typedef __attribute__((ext_vector_type(8))) float v8f;

#if __has_builtin(__builtin_amdgcn_global_load_async_to_lds_b128) && \
    __has_builtin(__builtin_amdgcn_global_load_async_to_lds_b32)  && \
    __has_builtin(__builtin_amdgcn_s_wait_asynccnt)
#define USE_ASYNC 1
// Builtin params are int4* / int* in the CUDA device/shared language address
// spaces; generic pointers of the matching pointee type convert implicitly
// (clang inserts the addrspacecast: generic->AS1 identity, generic->AS3
// low-32 truncation per the aperture mapping).
typedef int v4i_vs __attribute__((vector_size(4 * sizeof(int))));
#endif

// ---------------------------------------------------------------------------
// Kernel 1: streaming argmax + confusion-count histogram.
// ---------------------------------------------------------------------------
#if defined(USE_ASYNC)
// Issue exactly ASYNC_ITERS async b128 loads per lane (branch-free, clamped),
// so every wave's ASYNCcnt rises by exactly 13 per tile; plus a tiny b32 tail
// for the last (<16B) remainder of the whole array (none when N*51 % 4 == 0).
__device__ __forceinline__ void issue_tile_async(
    const float* __restrict__ pred, float4* ldsbuf, int tile, int numTiles,
    size_t totalF4, int rem)
{
  const int t = threadIdx.x;
  const v4i_vs* pred4 = (const v4i_vs*)pred;
  v4i_vs* lds4 = (v4i_vs*)ldsbuf;
  const size_t gbase = (size_t)tile * TILE_F4;
#pragma unroll
  for (int i = 0; i < ASYNC_ITERS; ++i) {
    const int lidx = i * THREADS + t;                       // 0 .. 1663
    size_t gidx = gbase + (size_t)lidx;
    gidx = (gidx < totalF4) ? gidx : (totalF4 - 1);         // stay inside pred
    __builtin_amdgcn_global_load_async_to_lds_b128(
        (v4i_vs*)(pred4 + gidx), lds4 + lidx, 0, 0);
  }
  if (rem && tile == numTiles - 1) {                        // generic tail
    const size_t remStart = totalF4 * 4;                    // global float idx
    const size_t off = remStart - (size_t)tile * TILE_FLOATS;  // float offset in tile
    int* ldsi = (int*)ldsbuf;
    for (int rr = t; rr < rem; rr += THREADS) {
      __builtin_amdgcn_global_load_async_to_lds_b32(
          (int*)(pred + remStart + rr), ldsi + off + rr, 0, 0);
    }
  }
}
#endif

__global__ __launch_bounds__(THREADS) void debias_hist_kernel(
    const float* __restrict__ pred, const int* __restrict__ gt,
    unsigned int* __restrict__ g_hist, int N, int numTiles)
{
  __shared__ float4       lds_buf[2][TILE_SLOTS];   // 2 x 26624 B
  __shared__ unsigned int lds_hist[HIST_BINS];      // 10404 B  (total 63652 B)
  const int t = threadIdx.x;

  for (int j = t; j < HIST_BINS; j += THREADS) lds_hist[j] = 0u;
  __syncthreads();   // publish hist zeroing before any atomics

#if defined(USE_ASYNC)
  const size_t totalF  = (size_t)N * NUM_REL;
  const size_t totalF4 = totalF >> 2;
  const int    rem     = (int)(totalF & 3);

  int tile = blockIdx.x;
  int cur  = 0;
  if (tile < numTiles)
    issue_tile_async(pred, lds_buf[0], tile, numTiles, totalF4, rem);

  for (; tile < numTiles; tile += gridDim.x) {
    const int next = tile + gridDim.x;
    if (next < numTiles) {
      // prefetch the tile after next while we're at it (global_prefetch_b8)
      const int pf = next + gridDim.x;
      if (pf < numTiles)
        __builtin_prefetch(pred + (size_t)pf * TILE_FLOATS + (size_t)t * 32, 0, 0);
      issue_tile_async(pred, lds_buf[cur ^ 1], next, numTiles, totalF4, rem);
      // 13 new ops in flight; waiting to <=13 drains the previous tile's loads
      // (async loads complete in order).
      __builtin_amdgcn_s_wait_asynccnt(13);
    } else {
      __builtin_amdgcn_s_wait_asynccnt(0);
    }
    __syncthreads();   // every wave drained its part of the current tile

    const int r = tile * TILE_ROWS + t;
    if (r < N) {
      const float* row = ((const float*)lds_buf[cur]) + t * NUM_REL;
      float best = row[1];
      int   bi   = 1;
#pragma unroll
      for (int c = 2; c < NUM_REL; ++c) {
        const float v = row[c];
        if (v > best) { best = v; bi = c; }
      }
      atomicAdd(&lds_hist[bi * NUM_REL + gt[r]], 1u);   // ds_add_u32
    }
    __syncthreads();   // all lanes done reading this buffer before it refills
    cur ^= 1;
  }
#else
  // Fallback: synchronous coalesced staging.
  for (int tile = blockIdx.x; tile < numTiles; tile += gridDim.x) {
    __syncthreads();
    const int nextTile = tile + gridDim.x;
    if (nextTile < numTiles)
      __builtin_prefetch(pred + (size_t)nextTile * TILE_FLOATS + (size_t)t * 32, 0, 0);
    const size_t base      = (size_t)tile * TILE_FLOATS;
    const int    validRows = min(TILE_ROWS, N - tile * TILE_ROWS);
    const int    validF    = validRows * NUM_REL;
    const int    nv4       = validF >> 2;
    const float4* __restrict__ src4 = (const float4*)(pred + base);
    float4* dst4 = (float4*)lds_buf[0];
    for (int i = t; i < nv4; i += THREADS) dst4[i] = src4[i];
    for (int i = (nv4 << 2) + t; i < validF; i += THREADS)
      ((float*)lds_buf[0])[i] = pred[base + i];
    __syncthreads();
    const int r = tile * TILE_ROWS + t;
    if (r < N) {
      const float* row = ((const float*)lds_buf[0]) + t * NUM_REL;
      float best = row[1];
      int   bi   = 1;
#pragma unroll
      for (int c = 2; c < NUM_REL; ++c) {
        const float v = row[c];
        if (v > best) { best = v; bi = c; }
      }
      atomicAdd(&lds_hist[bi * NUM_REL + gt[r]], 1u);
    }
  }
#endif

  __syncthreads();
  for (int j = t; j < HIST_BINS; j += THREADS) {
    const unsigned int v = lds_hist[j];
    if (v) atomicAdd(&g_hist[j], v);   // one global merge per block
  }
}

// ---------------------------------------------------------------------------
// Kernel 2: epilogue on the 51x51 matrix. One block, 128 threads (4 waves).
// Row sums via V_WMMA_F32_16X16X4_F32 against an all-ones B.
// ---------------------------------------------------------------------------
__global__ __launch_bounds__(128) void debias_epilogue_kernel(
    const unsigned int* __restrict__ g_hist,
    const float* __restrict__ rel_count,
    const int* __restrict__ istrain,
    float* __restrict__ out)
{
  __shared__ float histF[64 * PAD_COLS];  // zero-padded 64x52 counts
  __shared__ float S[64];                 // full row sums (incl. col 0)
  __shared__ float partial[128];
  __shared__ float s_total;

  const int t = threadIdx.x;

  if (istrain[0] == 0) {  // not training: identity pass-through
    for (int j = t; j < HIST_BINS; j += 128) out[j] = rel_count[j];
    return;
  }

  for (int j = t; j < 64 * PAD_COLS; j += 128) histF[j] = 0.0f;
  __syncthreads();
  for (int j = t; j < HIST_BINS; j += 128) {
    const int p = j / NUM_REL, g = j - p * NUM_REL;
    histF[p * PAD_COLS + g] = (float)g_hist[j];
  }

  // Deterministic fixed-order reduction of rel_count.sum() (==0 gate only).
  float acc = 0.0f;
  for (int j = t; j < HIST_BINS; j += 128) acc += rel_count[j];
  partial[t] = acc;
  __syncthreads();
  if (t == 0) {
    float s = 0.0f;
    for (int i = 0; i < 128; ++i) s += partial[i];
    s_total = s;
  }
  __syncthreads();

#if __has_builtin(__builtin_amdgcn_wmma_f32_16x16x4_f32)
  {
    const int lane = t & 31;
    const int wv   = t >> 5;                 // wave mt handles rows mt*16..mt*16+15
    const int m    = wv * 16 + (lane & 15);  // A 16x4 layout: lanes 0-15 K=0,1 ; 16-31 K=2,3
    const int koff = (lane >> 4) * 2;
    v2f b; b.x = 1.0f; b.y = 1.0f;           // all-ones B (4x16)
    v8f c = {};
    for (int kk = 0; kk < 13; ++kk) {        // 13 * K4 = 52 padded columns
      v2f a;
      a.x = histF[m * PAD_COLS + kk * 4 + koff];
      a.y = histF[m * PAD_COLS + kk * 4 + koff + 1];
      c = __builtin_amdgcn_wmma_f32_16x16x4_f32(
          /*neg_a=*/false, a, /*neg_b=*/false, b,
          /*c_mod=*/(short)0, c, /*reuse_a=*/false, /*reuse_b=*/false);
    }
    // D layout: VGPR d -> M=d (lanes 0-15) / M=d+8 (lanes 16-31); N = lane.
    if (lane == 0) {
#pragma unroll
      for (int d = 0; d < 8; ++d) S[wv * 16 + d] = c[d];
    }
    if (lane == 16) {
#pragma unroll
      for (int d = 0; d < 8; ++d) S[wv * 16 + 8 + d] = c[d];
    }
  }
#else
  for (int p = t; p < 64; p += 128) {
    float s = 0.0f;
    for (int g = 0; g < PAD_COLS; ++g) s += histF[p * PAD_COLS + g];
    S[p] = s;
  }
#endif
  __syncthreads();

  // gate uses row sum BEFORE zeroing column 0; norm denominator AFTER.
  const float s_rc = s_total;
  for (int j = t; j < HIST_BINS; j += 128) {
    const int p = j / NUM_REL, g = j - p * NUM_REL;
    const float h     = (g == 0) ? 0.0f : histF[p * PAD_COLS + g];
    const float denom = S[p] - histF[p * PAD_COLS + 0];
    const float norm  = h / (denom + 1e-10f);
    const float gate  = (S[p] > 0.0f) ? 0.9f : 1.0f;
    out[j] = (s_rc == 0.0f) ? norm : (gate * rel_count[j] + 0.1f * norm);
  }
}

// ---------------------------------------------------------------------------
extern "C" void kernel_launch(void* const* d_in, const int* in_sizes, int n_in,
                              void* d_out, int out_size, void* d_ws, size_t ws_size,
                              hipStream_t stream) {
  const float* pred      = (const float*)d_in[0];
  const float* rel_count = (const float*)d_in[1];
  const int*   gt        = (const int*)d_in[2];
  const int*   istrain   = (const int*)d_in[3];
  float*       out       = (float*)d_out;
  unsigned int* g_hist   = (unsigned int*)d_ws;

  const int N        = in_sizes[2];                       // 1,000,000 samples
  const int numTiles = (N + TILE_ROWS - 1) / TILE_ROWS;   // 7813

  (void)hipMemsetAsync(d_ws, 0, HIST_BINS * sizeof(unsigned int), stream);

  const int grid = numTiles < 640 ? numTiles : 640;       // persistent blocks
  debias_hist_kernel<<<grid, THREADS, 0, stream>>>(pred, gt, g_hist, N, numTiles);
  debias_epilogue_kernel<<<1, 128, 0, stream>>>(g_hist, rel_count, istrain, out);
}